// RGCN_Layer_45140106281569
// MI455X (gfx1250) — compile-verified
//
#include <hip/hip_runtime.h>
#include <hip/hip_bf16.h>

typedef __attribute__((ext_vector_type(2))) float v2f;
typedef __attribute__((ext_vector_type(8))) float v8f;

constexpr int N_   = 50000;
constexpr int R_   = 8;
constexpr int E_   = 100000;
constexpr int B_   = 16384;
constexpr int D_   = 128;
constexpr int OUT_ = 128;
constexpr int LDSP = 130;                 // padded LDS row stride (bank-safe)

// ---------------------------------------------------------------------------
// Cooperative LDS stage with fused transpose:  smem[o*130 + k] = W[k*128 + o]
// Coalesced global reads; LDS write bank = (2*o + k) % 64 -> conflict-free.
// ---------------------------------------------------------------------------
__device__ __forceinline__ void stage_weights_T(const float* __restrict__ W,
                                                float* __restrict__ smem) {
    for (int i = threadIdx.x; i < D_ * OUT_; i += blockDim.x) {
        const int k = i >> 7;
        const int o = i & 127;
        smem[o * LDSP + k] = W[i];
    }
    __syncthreads();
}

// ---------------------------------------------------------------------------
// WMMA inner loop: one wave computes a 16-row x 128-col f32 GEMM strip.
//   acc[c] += A(16 x 128) * W(128 x 128, col tile c), V_WMMA_F32_16X16X4_F32.
// A from global (b64 per lane), B from LDS (ds_load_b64 per lane).
// A-layout : lane holds X [m          , k + 2*(lane>>4) + {0,1}], m = lane&15
// B-layout : lane holds Wt[c*16 + m   , k + 2*(lane>>4) + {0,1}] (smem, o-major)
// Loads and WMMAs are phase-separated so B operands occupy 8 distinct
// register pairs (avoids the serialized load->wait0->wmma chain).
// ---------------------------------------------------------------------------
__device__ __forceinline__ void wmma_strip_lds(const float* __restrict__ arow,
                                               const float* __restrict__ smem,
                                               int lane, v8f acc[8]) {
    const int m    = lane & 15;
    const int koff = (lane >> 4) * 2;
    const float* __restrict__ bb = smem + m * LDSP + koff;
    for (int k = 0; k < D_; k += 4) {
        const v2f a = *(const v2f*)(arow + k + koff);
        v2f b[8];
#pragma unroll
        for (int c = 0; c < 8; ++c)
            b[c] = *(const v2f*)(bb + c * 16 * LDSP + k);
#pragma unroll
        for (int c = 0; c < 8; ++c)
            acc[c] = __builtin_amdgcn_wmma_f32_16x16x4_f32(
                false, a, false, b[c], (short)0, acc[c], false, false);
    }
}

// ---------------------------------------------------------------------------
// P = emb @ K_r   (N x OUT); one wave per 16-row strip (3125 strips).
// Idle tail waves clamp to row 0 and skip the store (wave-uniform; EXEC
// stays all-ones through every WMMA; all waves reach the barrier).
// ---------------------------------------------------------------------------
__global__ void rgcn_proj_kernel(const float* __restrict__ emb,
                                 const float* __restrict__ Wr,
                                 float* __restrict__ P) {
    __shared__ float smem[OUT_ * LDSP];
    stage_weights_T(Wr, smem);

    const int  lane   = threadIdx.x & 31;
    const int  wid    = blockIdx.x * (blockDim.x >> 5) + (threadIdx.x >> 5);
    const bool active = (wid < N_ / 16);
    const int  row0   = active ? wid * 16 : 0;
    const int  m      = lane & 15;
    const int  khalf  = lane >> 4;

    const float* arow = emb + (size_t)(row0 + m) * D_;
    v8f acc[8] = {};
    wmma_strip_lds(arow, smem, lane, acc);

    if (active) {
#pragma unroll
        for (int v = 0; v < 8; ++v) {
            const size_t row = (size_t)(row0 + v + 8 * khalf);
#pragma unroll
            for (int c = 0; c < 8; ++c)
                P[row * OUT_ + c * 16 + m] = acc[c][v];
        }
    }
}

// ---------------------------------------------------------------------------
// Edge scatter: acc[rows[e], :] += vals[e] * P[cols[e], :]
// One wave per edge; lane handles 4 contiguous floats (b128 load, 4x native
// f32 atomic adds into the 25.6MB L2-resident accumulator).
// ---------------------------------------------------------------------------
__global__ void rgcn_scatter_kernel(const int* __restrict__ rows,
                                    const int* __restrict__ cols,
                                    const float* __restrict__ vals,
                                    const float* __restrict__ P,
                                    float* __restrict__ accbuf) {
    const int lane = threadIdx.x & 31;
    const int e    = blockIdx.x * (blockDim.x >> 5) + (threadIdx.x >> 5);
    if (e >= E_) return;

    const int   rr = rows[e];
    const int   cc = cols[e];
    const float vv = vals[e];

    const float4 p = ((const float4*)(P + (size_t)cc * OUT_))[lane];
    float* dst = accbuf + (size_t)rr * OUT_ + lane * 4;
    __hip_atomic_fetch_add(dst + 0, vv * p.x, __ATOMIC_RELAXED, __HIP_MEMORY_SCOPE_AGENT);
    __hip_atomic_fetch_add(dst + 1, vv * p.y, __ATOMIC_RELAXED, __HIP_MEMORY_SCOPE_AGENT);
    __hip_atomic_fetch_add(dst + 2, vv * p.z, __ATOMIC_RELAXED, __HIP_MEMORY_SCOPE_AGENT);
    __hip_atomic_fetch_add(dst + 3, vv * p.w, __ATOMIC_RELAXED, __HIP_MEMORY_SCOPE_AGENT);
}

// ---------------------------------------------------------------------------
// out[b,:] = sigmoid( X[b,:] @ W_self + acc[idx[b], :] )
// 1024 strips / 128 blocks exactly; no tail handling needed.
// ---------------------------------------------------------------------------
__global__ void rgcn_final_kernel(const float* __restrict__ Xe,
                                  const int* __restrict__ idx,
                                  const float* __restrict__ selfK,
                                  const float* __restrict__ accbuf,
                                  float* __restrict__ out) {
    __shared__ float smem[OUT_ * LDSP];
    stage_weights_T(selfK, smem);

    const int lane  = threadIdx.x & 31;
    const int wid   = blockIdx.x * (blockDim.x >> 5) + (threadIdx.x >> 5);
    const int row0  = wid * 16;
    const int m     = lane & 15;
    const int khalf = lane >> 4;

    const float* arow = Xe + (size_t)(row0 + m) * D_;
    v8f acc[8] = {};
    wmma_strip_lds(arow, smem, lane, acc);

#pragma unroll
    for (int v = 0; v < 8; ++v) {
        const int    row   = row0 + v + 8 * khalf;
        const size_t gbase = (size_t)idx[row] * OUT_;
#pragma unroll
        for (int c = 0; c < 8; ++c) {
            const float x = acc[c][v] + accbuf[gbase + c * 16 + m];
            out[(size_t)row * OUT_ + c * 16 + m] = 1.0f / (1.0f + __expf(-x));
        }
    }
}

// ---------------------------------------------------------------------------
// Zero the N x OUT accumulator (float4 stores)
// ---------------------------------------------------------------------------
__global__ void rgcn_zero_kernel(float4* __restrict__ p, int n4) {
    const int i = blockIdx.x * blockDim.x + threadIdx.x;
    if (i < n4) p[i] = make_float4(0.0f, 0.0f, 0.0f, 0.0f);
}

extern "C" void kernel_launch(void* const* d_in, const int* in_sizes, int n_in,
                              void* d_out, int out_size, void* d_ws, size_t ws_size,
                              hipStream_t stream) {
    const float* emb      = (const float*)d_in[0];
    const int*   head_idx = (const int*)  d_in[1];
    const float* head_e   = (const float*)d_in[2];
    const int*   tail_idx = (const int*)  d_in[3];
    const float* tail_e   = (const float*)d_in[4];
    const int*   adj_rows = (const int*)  d_in[5];
    const int*   adj_cols = (const int*)  d_in[6];
    const float* adj_vals = (const float*)d_in[7];
    const float* relK     = (const float*)d_in[8];
    const float* selfK    = (const float*)d_in[9];
    float*       out      = (float*)d_out;

    // workspace: [acc : N*OUT floats][P : N*OUT floats]  ~51.2 MB
    float* accbuf = (float*)d_ws;
    float* P      = accbuf + (size_t)N_ * OUT_;

    // 1) zero the L2-resident accumulator
    const int n4 = (N_ * OUT_) / 4;
    rgcn_zero_kernel<<<(n4 + 255) / 256, 256, 0, stream>>>((float4*)accbuf, n4);

    // 2) per relation: dense WMMA projection, then edge scatter-accumulate
    const int projBlocks = (N_ / 16 + 7) / 8;   // 8 waves per 256-thread block
    const int scatBlocks = (E_ + 7) / 8;
    for (int r = 0; r < R_; ++r) {
        rgcn_proj_kernel<<<projBlocks, 256, 0, stream>>>(
            emb, relK + (size_t)r * D_ * OUT_, P);
        rgcn_scatter_kernel<<<scatBlocks, 256, 0, stream>>>(
            adj_rows + (size_t)r * E_, adj_cols + (size_t)r * E_,
            adj_vals + (size_t)r * E_, P, accbuf);
    }

    // 3) self-kernel GEMM + gathered accumulator + sigmoid
    const int finBlocks = (B_ / 16) / 8;        // exact: 128 blocks
    rgcn_final_kernel<<<finBlocks, 256, 0, stream>>>(head_e, head_idx, selfK, accbuf, out);
    rgcn_final_kernel<<<finBlocks, 256, 0, stream>>>(tail_e, tail_idx, selfK, accbuf,
                                                     out + (size_t)B_ * OUT_);
}